// AttributeImageCaptioner_37744172598013
// MI455X (gfx1250) — compile-verified
//
#include <hip/hip_runtime.h>
#include <hip/hip_bf16.h>
#include <math.h>

#define BB    32
#define TT    64
#define KK    5
#define VOC   32000
#define DE    300
#define DA    300
#define DI    2048
#define HH    512
#define DF    (DI + DA)          // 2348 (image + attr features)
#define G4H   (4 * HH)           // 2048 gate width
// lstm_kernel row layout: [0,DE) word emb | [DE,DE+DF) feats | [DE+DF, DE+DF+HH) hidden

typedef __attribute__((ext_vector_type(2))) float v2f;
typedef __attribute__((ext_vector_type(8))) float v8f;

__device__ __forceinline__ v8f wmma_k4(v2f a, v2f b, v8f c) {
    // D = A(16x4,f32) x B(4x16,f32) + C(16x16,f32)
    return __builtin_amdgcn_wmma_f32_16x16x4_f32(false, a, false, b, (short)0, c, false, false);
}

__device__ __forceinline__ float sigf(float x) { return 1.0f / (1.0f + expf(-x)); }

// monotone key: fkey(a) >= fkey(b)  <=>  a >= b  (for non-NaN floats)
__device__ __forceinline__ unsigned fkey(float v) {
    unsigned u = __float_as_uint(v);
    return (u & 0x80000000u) ? ~u : (u | 0x80000000u);
}

// ---------------------------------------------------------------------------
// 1) feats[b, 0:DI) = image features ; feats[b, DI:DF) = mean_k attr_emb
// ---------------------------------------------------------------------------
__global__ void feats_kernel(const int* __restrict__ topk,
                             const float* __restrict__ img,
                             const float* __restrict__ aemb,
                             float* __restrict__ feats) {
    int idx = blockIdx.x * blockDim.x + threadIdx.x;
    if (idx >= BB * DF) return;
    int b = idx / DF, d = idx % DF;
    float v;
    if (d < DI) {
        v = img[b * DI + d];
    } else {
        int da = d - DI;
        float s = 0.0f;
#pragma unroll
        for (int k = 0; k < KK; ++k)
            s += aemb[topk[b * KK + k] * DA + da];
        v = s * (1.0f / (float)KK);
    }
    feats[idx] = v;
}

// ---------------------------------------------------------------------------
// 2) init: zero h0, packed argmax keys, and the grid-barrier counter
// ---------------------------------------------------------------------------
__global__ void init_kernel(float* __restrict__ h0,
                            unsigned long long* __restrict__ rowkey,
                            unsigned* __restrict__ barrier_cnt) {
    int i = blockIdx.x * blockDim.x + threadIdx.x;
    if (i < BB * HH) h0[i] = 0.0f;
    if (i < BB * TT) rowkey[i] = 0ull;
    if (i == 0) *barrier_cnt = 0u;
}

// ---------------------------------------------------------------------------
// 3) zfeat[32,2048] = feats[32,2348] @ W[DE:DE+DF,:] + bias  (once, not per-t)
//    2 waves/block: wave = one 16-row m-tile, 4 n-tiles (32 acc VGPRs/wave)
// ---------------------------------------------------------------------------
__global__ void zfeat_kernel(const float* __restrict__ feats,
                             const float* __restrict__ Wk,
                             const float* __restrict__ bias,
                             float* __restrict__ zfeat) {
    const int lane = threadIdx.x & 31;
    const int wid  = threadIdx.x >> 5;          // m-tile
    const int n0   = blockIdx.x * 64;
    const int kOff = (lane >> 4) * 2;           // lanes 16-31 hold K=2,3
    const int nl   = lane & 15;
    const int mrow = wid * 16 + nl;             // A row for this lane

    v8f acc[4];
#pragma unroll
    for (int nt = 0; nt < 4; ++nt) acc[nt] = (v8f)0.0f;

    for (int k0 = 0; k0 < DF; k0 += 4) {
        v2f a = *(const v2f*)&feats[(size_t)mrow * DF + k0 + kOff];
#pragma unroll
        for (int nt = 0; nt < 4; ++nt) {
            int n = n0 + nt * 16 + nl;
            const float* wp = &Wk[(size_t)(DE + k0 + kOff) * G4H + n];
            v2f bf; bf.x = wp[0]; bf.y = wp[G4H];
            acc[nt] = wmma_k4(a, bf, acc[nt]);
        }
    }
#pragma unroll
    for (int nt = 0; nt < 4; ++nt) {
        int n = n0 + nt * 16 + nl;
        float bv = bias[n];
#pragma unroll
        for (int r = 0; r < 8; ++r) {
            int row = wid * 16 + r + ((lane >> 4) << 3);
            zfeat[(size_t)row * G4H + n] = acc[nt][r] + bv;
        }
    }
}

// ---------------------------------------------------------------------------
// 4) zpre[b*T+t, :] = word_emb[seq[b,t]] @ W[0:DE, :] + zfeat[b, :]
//    1 wave/block computes 16 rows x 64 cols, K=300  (32 acc VGPRs)
// ---------------------------------------------------------------------------
__global__ void zword_kernel(const int* __restrict__ seq,
                             const float* __restrict__ wemb,
                             const float* __restrict__ Wk,
                             const float* __restrict__ zfeat,
                             float* __restrict__ zpre) {
    const int lane = threadIdx.x;
    const int m0   = blockIdx.x * 16;
    const int n0   = blockIdx.y * 64;
    const int kOff = (lane >> 4) * 2;
    const int nl   = lane & 15;

    const int widx = seq[m0 + nl];                 // row -> word index
    const float* arow = &wemb[(size_t)widx * DE];

    v8f acc[4];
#pragma unroll
    for (int nt = 0; nt < 4; ++nt) acc[nt] = (v8f)0.0f;

    for (int k0 = 0; k0 < DE; k0 += 4) {           // 300 = 75 * 4
        v2f a = *(const v2f*)&arow[k0 + kOff];
#pragma unroll
        for (int nt = 0; nt < 4; ++nt) {
            int n = n0 + nt * 16 + nl;
            const float* wp = &Wk[(size_t)(k0 + kOff) * G4H + n];
            v2f bf; bf.x = wp[0]; bf.y = wp[G4H];
            acc[nt] = wmma_k4(a, bf, acc[nt]);
        }
    }
#pragma unroll
    for (int nt = 0; nt < 4; ++nt) {
        int n = n0 + nt * 16 + nl;
#pragma unroll
        for (int r = 0; r < 8; ++r) {
            int row = m0 + r + ((lane >> 4) << 3);          // row = b*T + t
            zpre[(size_t)row * G4H + n] = acc[nt][r] + zfeat[(size_t)(row >> 6) * G4H + n];
        }
    }
}

// ---------------------------------------------------------------------------
// 5) PERSISTENT LSTM: one launch for all 64 steps.
//    32 blocks x 2 waves; block owns hidden slice j0..j0+15 (all 4 gates).
//    - W_h slice (512x64 = 128 KB) staged into LDS once (320 KB/WGP on CDNA5)
//    - cell state c lives in registers (fixed lane ownership across steps)
//    - grid-wide counter barrier between steps (counter zeroed per call)
// ---------------------------------------------------------------------------
__global__ void lstm_persistent_kernel(const float* __restrict__ zpre,
                                       const float* __restrict__ Wk,
                                       const int* __restrict__ lens,
                                       float* __restrict__ hbuf,     // 2*B*H ping-pong
                                       float* __restrict__ hseq,
                                       unsigned* __restrict__ barrier_cnt) {
    extern __shared__ float Wh[];               // [HH][64]
    const int tid  = threadIdx.x;
    const int lane = tid & 31;
    const int wid  = tid >> 5;                  // m-tile (batch rows 16*wid..)
    const int j0   = blockIdx.x * 16;
    const int kOff = (lane >> 4) * 2;
    const int nl   = lane & 15;
    const int arow = wid * 16 + nl;             // batch row for A fragment

    // stage W_h slice: column c of LDS tile = gate (c>>4), hidden j0+(c&15)
    {
        const int g = tid >> 4, j = j0 + (tid & 15);
        for (int k = 0; k < HH; ++k)
            Wh[k * 64 + tid] = Wk[(size_t)(DE + DF + k) * G4H + g * HH + j];
    }
    __syncthreads();

    float creg[8];                              // persistent cell state
#pragma unroll
    for (int r = 0; r < 8; ++r) creg[r] = 0.0f;
    int lenr[8];
#pragma unroll
    for (int r = 0; r < 8; ++r)
        lenr[r] = lens[wid * 16 + r + ((lane >> 4) << 3)];

    for (int t = 0; t < TT; ++t) {
        const float* hin = hbuf + (size_t)(t & 1) * BB * HH;
        float* hout      = hbuf + (size_t)((t + 1) & 1) * BB * HH;

        v8f acc[4];
#pragma unroll
        for (int g = 0; g < 4; ++g)
#pragma unroll
            for (int r = 0; r < 8; ++r) {
                int b = wid * 16 + r + ((lane >> 4) << 3);
                acc[g][r] = zpre[(size_t)(b * TT + t) * G4H + g * HH + j0 + nl];
            }

        for (int k0 = 0; k0 < HH; k0 += 4) {
            v2f a = *(const v2f*)&hin[(size_t)arow * HH + k0 + kOff];
#pragma unroll
            for (int g = 0; g < 4; ++g) {
                int c = g * 16 + nl;
                v2f bf;
                bf.x = Wh[(k0 + kOff) * 64 + c];
                bf.y = Wh[(k0 + kOff + 1) * 64 + c];
                acc[g] = wmma_k4(a, bf, acc[g]);
            }
        }

#pragma unroll
        for (int r = 0; r < 8; ++r) {
            int b = wid * 16 + r + ((lane >> 4) << 3);
            int j = j0 + nl;
            float zi = acc[0][r], zf = acc[1][r], zg = acc[2][r], zo = acc[3][r];
            float cn = sigf(zf) * creg[r] + sigf(zi) * tanhf(zg);
            creg[r] = cn;
            float hn = sigf(zo) * tanhf(cn);
            hout[(size_t)b * HH + j] = hn;
            hseq[(size_t)(b * TT + t) * HH + j] = (t < lenr[r]) ? hn : 0.0f;
        }

        // grid-wide barrier: all blocks must finish step t before t+1 reads h
        __syncthreads();
        if (tid == 0) {
            __hip_atomic_fetch_add(barrier_cnt, 1u, __ATOMIC_RELEASE,
                                   __HIP_MEMORY_SCOPE_AGENT);
            const unsigned target = (unsigned)gridDim.x * (unsigned)(t + 1);
            while (__hip_atomic_load(barrier_cnt, __ATOMIC_ACQUIRE,
                                     __HIP_MEMORY_SCOPE_AGENT) < target)
                __builtin_amdgcn_s_sleep(2);
        }
        __syncthreads();
        __threadfence();
    }
}

// ---------------------------------------------------------------------------
// 6) logits[2048,32000] = hseq @ W_logits + b_logits, with FUSED partial
//    argmax (packed-key atomicMax per row) to avoid re-reading 262 MB.
//    grid (64, 500) x 2 waves; wave = one 16-row m-tile, 4 n-tiles.
// ---------------------------------------------------------------------------
__global__ void logits_kernel(const float* __restrict__ hseq,
                              const float* __restrict__ Wl,
                              const float* __restrict__ bl,
                              float* __restrict__ out,
                              unsigned long long* __restrict__ rowkey) {
    const int lane = threadIdx.x & 31;
    const int wid  = threadIdx.x >> 5;
    const int m0   = blockIdx.x * 32 + wid * 16;
    const int n0   = blockIdx.y * 64;
    const int kOff = (lane >> 4) * 2;
    const int nl   = lane & 15;

    v8f acc[4];
#pragma unroll
    for (int nt = 0; nt < 4; ++nt) {
        float bv = bl[n0 + nt * 16 + nl];
#pragma unroll
        for (int r = 0; r < 8; ++r) acc[nt][r] = bv;
    }

    for (int k0 = 0; k0 < HH; k0 += 4) {
        v2f a = *(const v2f*)&hseq[(size_t)(m0 + nl) * HH + k0 + kOff];
#pragma unroll
        for (int nt = 0; nt < 4; ++nt) {
            int n = n0 + nt * 16 + nl;
            const float* wp = &Wl[(size_t)(k0 + kOff) * VOC + n];
            v2f bf; bf.x = wp[0]; bf.y = wp[VOC];
            acc[nt] = wmma_k4(a, bf, acc[nt]);
        }
    }

    // store logits tile
#pragma unroll
    for (int nt = 0; nt < 4; ++nt) {
        int n = n0 + nt * 16 + nl;
#pragma unroll
        for (int r = 0; r < 8; ++r) {
            int row = m0 + r + ((lane >> 4) << 3);
            out[(size_t)row * VOC + n] = acc[nt][r];
        }
    }

    // fused per-row argmax over this 64-column tile.
    // C striping: for fixed r, the 16 lanes of each half-wave hold 16
    // consecutive columns of ONE row -> shfl_xor reduce within the half.
#pragma unroll
    for (int r = 0; r < 8; ++r) {
        float bv = acc[0][r];
        int   bn = n0 + nl;
#pragma unroll
        for (int nt = 1; nt < 4; ++nt) {
            float v = acc[nt][r];
            int   n = n0 + nt * 16 + nl;
            if (v > bv || (v == bv && n < bn)) { bv = v; bn = n; }
        }
#pragma unroll
        for (int m = 1; m <= 8; m <<= 1) {      // stays within 16-lane half
            float ov = __shfl_xor(bv, m, 32);
            int   on = __shfl_xor(bn, m, 32);
            if (ov > bv || (ov == bv && on < bn)) { bv = ov; bn = on; }
        }
        if ((lane & 15) == 0) {
            int row = m0 + r + ((lane >> 4) << 3);
            unsigned long long key =
                ((unsigned long long)fkey(bv) << 32) | (unsigned)(~(unsigned)bn);
            atomicMax(&rowkey[row], key);       // ties -> max(~n) -> min n
        }
    }
}

// ---------------------------------------------------------------------------
// 7) decode packed keys -> int32 argmax indices at tail of d_out
// ---------------------------------------------------------------------------
__global__ void argmax_decode_kernel(const unsigned long long* __restrict__ rowkey,
                                     int* __restrict__ outi) {
    int row = blockIdx.x * blockDim.x + threadIdx.x;
    if (row < BB * TT)
        outi[row] = (int)(~(unsigned)(rowkey[row] & 0xFFFFFFFFull));
}

// ---------------------------------------------------------------------------
extern "C" void kernel_launch(void* const* d_in, const int* in_sizes, int n_in,
                              void* d_out, int out_size, void* d_ws, size_t ws_size,
                              hipStream_t stream) {
    const int*   topk = (const int*)d_in[0];
    const float* img  = (const float*)d_in[1];
    const int*   seq  = (const int*)d_in[2];
    const int*   lens = (const int*)d_in[3];
    const float* wemb = (const float*)d_in[4];
    const float* aemb = (const float*)d_in[5];
    const float* Wk   = (const float*)d_in[6];
    const float* bk   = (const float*)d_in[7];
    const float* Wl   = (const float*)d_in[8];
    const float* bl   = (const float*)d_in[9];

    // workspace layout: 8-byte-aligned keys + barrier counter first
    unsigned long long* rowkey = (unsigned long long*)d_ws;
    unsigned* barrier_cnt = (unsigned*)(rowkey + BB * TT);
    float* ws     = (float*)(rowkey + BB * TT + 1);
    float* feats  = ws;                 size_t o = (size_t)BB * DF;
    float* zfeat  = ws + o;             o += (size_t)BB * G4H;
    float* zpre   = ws + o;             o += (size_t)BB * TT * G4H;
    float* hbuf   = ws + o;             o += (size_t)2 * BB * HH;   // ping-pong
    float* hseq   = ws + o;             o += (size_t)BB * TT * HH;

    float* logits = (float*)d_out;
    int*   amax   = (int*)(logits + (size_t)BB * TT * VOC);

    feats_kernel<<<(BB * DF + 255) / 256, 256, 0, stream>>>(topk, img, aemb, feats);
    init_kernel<<<(BB * HH + 255) / 256, 256, 0, stream>>>(hbuf, rowkey, barrier_cnt);
    zfeat_kernel<<<G4H / 64, 64, 0, stream>>>(feats, Wk, bk, zfeat);
    zword_kernel<<<dim3(BB * TT / 16, G4H / 64), 32, 0, stream>>>(seq, wemb, Wk, zfeat, zpre);

    // one persistent launch for the whole recurrence; 128 KB LDS per block
    lstm_persistent_kernel<<<HH / 16, 64, (size_t)HH * 64 * sizeof(float), stream>>>(
        zpre, Wk, lens, hbuf, hseq, barrier_cnt);

    logits_kernel<<<dim3(BB * TT / 32, VOC / 64), 64, 0, stream>>>(hseq, Wl, bl, logits, rowkey);
    argmax_decode_kernel<<<(BB * TT + 255) / 256, 256, 0, stream>>>(rowkey, amax);
}